// Goal_example_model_12171937317316
// MI455X (gfx1250) — compile-verified
//
#include <hip/hip_runtime.h>
#include <math.h>

typedef __attribute__((ext_vector_type(2))) float v2f;
typedef __attribute__((ext_vector_type(8))) float v8f;

#define NB   4096      // batch (agents)
#define NT   8         // timesteps
#define PREDL 12

__device__ __forceinline__ float sigm_(float x) { return 1.0f / (1.0f + __expf(-x)); }

// ---------------------------------------------------------------------------
// Fused traj_enc layer1+layer2:  H2 = relu( relu(X@W1^T + b1) @ W2^T + b2 )
// X:(32768,2)  W1:(512,2) b1:(512)  W2:(256,512) b2:(256)  H2:(32768,256)
// One wave computes a 16x16 tile with V_WMMA_F32_16X16X4_F32; the 16x4 A
// fragment is recomputed on the fly from X (2 FMAs/elt) -> no H1 in HBM.
// ---------------------------------------------------------------------------
__global__ void __launch_bounds__(256)
k_enc12(const float* __restrict__ X, const float* __restrict__ W1,
        const float* __restrict__ b1, const float* __restrict__ W2,
        const float* __restrict__ b2, float* __restrict__ H2)
{
    const int K = 512, N = 256;
    int wave = threadIdx.x >> 5;
    int lane = threadIdx.x & 31;
    int tn   = blockIdx.x & 15;                 // 16 column tiles
    int tm   = ((blockIdx.x >> 4) << 3) + wave; // 2048 row tiles
    int grp  = lane >> 4;
    int m    = tm * 16 + (lane & 15);
    int n    = tn * 16 + (lane & 15);

    float x0 = X[m * 2 + 0], x1 = X[m * 2 + 1];
    v8f acc = {0.f, 0.f, 0.f, 0.f, 0.f, 0.f, 0.f, 0.f};

    for (int k0 = 0; k0 < K; k0 += 4) {
        int ka = k0 + 2 * grp;
        v2f a, b;
        float t0 = fmaf(x0, W1[ka * 2 + 0], fmaf(x1, W1[ka * 2 + 1], b1[ka]));
        float t1 = fmaf(x0, W1[ka * 2 + 2], fmaf(x1, W1[ka * 2 + 3], b1[ka + 1]));
        a.x = fmaxf(t0, 0.f);
        a.y = fmaxf(t1, 0.f);
        b.x = W2[n * K + ka];
        b.y = W2[n * K + ka + 1];
        acc = __builtin_amdgcn_wmma_f32_16x16x4_f32(false, a, false, b,
                                                    (short)0, acc, false, false);
    }
    int col = tn * 16 + (lane & 15);
    float bb = b2[col];
#pragma unroll
    for (int r = 0; r < 8; ++r) {
        int row = tm * 16 + r + 8 * grp;
        H2[row * N + col] = fmaxf(acc[r] + bb, 0.f);
    }
}

// ---------------------------------------------------------------------------
// Generic WMMA f32 GEMM: C = act(A(M,K) @ W(N,K)^T + bias)
// act: 0 = none, 1 = PReLU with *slope (slope==nullptr -> 0 -> ReLU)
// M multiple of 16 and M/16 multiple of 8 (true for all call sites).
// N, K arbitrary (zero-padded loads, masked stores).
// ---------------------------------------------------------------------------
__global__ void __launch_bounds__(256)
k_gemm(const float* __restrict__ A, const float* __restrict__ W,
       const float* __restrict__ bias, const float* __restrict__ slope,
       float* __restrict__ C, int M, int N, int K, int act)
{
    int ntn  = (N + 15) >> 4;
    int wave = threadIdx.x >> 5;
    int lane = threadIdx.x & 31;
    int tn   = blockIdx.x % ntn;
    int tm   = (blockIdx.x / ntn) * 8 + wave;
    int grp  = lane >> 4;
    int m    = tm * 16 + (lane & 15);
    int n    = tn * 16 + (lane & 15);
    bool nok = (n < N);

    v8f acc = {0.f, 0.f, 0.f, 0.f, 0.f, 0.f, 0.f, 0.f};
    for (int k0 = 0; k0 < K; k0 += 4) {
        int ka = k0 + 2 * grp;
        v2f a, b;
        a.x = (ka < K)            ? A[m * K + ka]     : 0.f;
        a.y = (ka + 1 < K)        ? A[m * K + ka + 1] : 0.f;
        b.x = (nok && ka < K)     ? W[n * K + ka]     : 0.f;
        b.y = (nok && ka + 1 < K) ? W[n * K + ka + 1] : 0.f;
        acc = __builtin_amdgcn_wmma_f32_16x16x4_f32(false, a, false, b,
                                                    (short)0, acc, false, false);
    }
    if (nok) {
        float bb = bias ? bias[n] : 0.f;
        float p  = (act && slope) ? *slope : 0.f;
#pragma unroll
        for (int r = 0; r < 8; ++r) {
            int row = tm * 16 + r + 8 * grp;
            float v = acc[r] + bb;
            if (act) v = (v >= 0.f) ? v : p * v;
            C[row * N + n] = v;
        }
    }
}

// ---------------------------------------------------------------------------
// Encoder LSTM (hidden 5) over T=8, one thread per agent, weights in LDS.
// ---------------------------------------------------------------------------
__global__ void __launch_bounds__(256)
k_enc_lstm(const float* __restrict__ feats, const float* __restrict__ Wih,
           const float* __restrict__ Whh, const float* __restrict__ bih,
           const float* __restrict__ bhh, float* __restrict__ h_out)
{
    __shared__ float sWih[100], sWhh[100], sb[20];
    int tid = threadIdx.x;
    if (tid < 100) { sWih[tid] = Wih[tid]; sWhh[tid] = Whh[tid]; }
    if (tid < 20)  sb[tid] = bih[tid] + bhh[tid];
    __syncthreads();

    int b = blockIdx.x * 256 + tid;
    float h[5] = {0, 0, 0, 0, 0}, c[5] = {0, 0, 0, 0, 0};
    for (int t = 0; t < NT; ++t) {
        float x[5];
#pragma unroll
        for (int k = 0; k < 5; ++k) x[k] = feats[(b * NT + t) * 5 + k];
        float gates[20];
#pragma unroll
        for (int j = 0; j < 20; ++j) {
            float s = sb[j];
#pragma unroll
            for (int k = 0; k < 5; ++k)
                s = fmaf(x[k], sWih[j * 5 + k], fmaf(h[k], sWhh[j * 5 + k], s));
            gates[j] = s;
        }
#pragma unroll
        for (int k = 0; k < 5; ++k) {
            float ig = sigm_(gates[k]);
            float fg = sigm_(gates[5 + k]);
            float gg = tanhf(gates[10 + k]);
            float og = sigm_(gates[15 + k]);
            c[k] = fmaf(fg, c[k], ig * gg);
            h[k] = og * tanhf(c[k]);
        }
    }
#pragma unroll
    for (int k = 0; k < 5; ++k) h_out[b * 5 + k] = h[k];
}

// ---------------------------------------------------------------------------
// Fused non-local attention row pass (softmax denominator cancels under the
// L1 renormalization):  h_inp[row] += (Σ_j e^{s_j-m}·mask_j·g_j) / Σ_j e^{s_j-m}·mask_j
// One 256-thread block per row; single streaming read of the mask row.
// ---------------------------------------------------------------------------
__global__ void __launch_bounds__(256)
k_attention(const float* __restrict__ theta, const float* __restrict__ phi,
            const float* __restrict__ g, const float* __restrict__ mask,
            float* __restrict__ h_inp)
{
    __shared__ float sred[6][256];
    int row = blockIdx.x, tid = threadIdx.x;
    float tr[5];
#pragma unroll
    for (int k = 0; k < 5; ++k) tr[k] = theta[row * 5 + k];

    // pass 1: row max of s
    float mx = -3.4e38f;
    for (int j = tid; j < NB; j += 256) {
        float s = 0.f;
#pragma unroll
        for (int k = 0; k < 5; ++k) s = fmaf(tr[k], phi[j * 5 + k], s);
        mx = fmaxf(mx, s);
    }
    sred[0][tid] = mx;
    __syncthreads();
    for (int off = 128; off > 0; off >>= 1) {
        if (tid < off) sred[0][tid] = fmaxf(sred[0][tid], sred[0][tid + off]);
        __syncthreads();
    }
    mx = sred[0][0];
    __syncthreads();

    // pass 2: masked exp-weighted sum of g and partition
    float z = 0.f, acc[5] = {0, 0, 0, 0, 0};
    const float* mrow = mask + (size_t)row * NB;
    for (int j = tid; j < NB; j += 256) {
        float s = 0.f;
#pragma unroll
        for (int k = 0; k < 5; ++k) s = fmaf(tr[k], phi[j * 5 + k], s);
        float e = __expf(s - mx) * mrow[j];
        z += e;
#pragma unroll
        for (int k = 0; k < 5; ++k) acc[k] = fmaf(e, g[j * 5 + k], acc[k]);
    }
    sred[0][tid] = z;
#pragma unroll
    for (int k = 0; k < 5; ++k) sred[k + 1][tid] = acc[k];
    __syncthreads();
    for (int off = 128; off > 0; off >>= 1) {
        if (tid < off)
#pragma unroll
            for (int q = 0; q < 6; ++q) sred[q][tid] += sred[q][tid + off];
        __syncthreads();
    }
    if (tid < 5) {
        float zz = fmaxf(sred[0][0], 1e-30f);
        h_inp[row * 5 + tid] = sred[tid + 1][0] / zz + h_inp[row * 5 + tid];
    }
}

// ---------------------------------------------------------------------------
// Decoder init: a = v[:, -1, :2]; h=c=0; also writes a_pred[0].
// ---------------------------------------------------------------------------
__global__ void __launch_bounds__(256)
k_dec_init(const float* __restrict__ v, float* __restrict__ a,
           float* __restrict__ h, float* __restrict__ c,
           float* __restrict__ apred0)
{
    int b = blockIdx.x * 256 + threadIdx.x;
    float a0 = v[(b * NT + 7) * 2 + 0];
    float a1 = v[(b * NT + 7) * 2 + 1];
    a[b * 2] = a0; a[b * 2 + 1] = a1;
    apred0[b * 2] = a0; apred0[b * 2 + 1] = a1;
#pragma unroll
    for (int k = 0; k < 5; ++k) { h[b * 5 + k] = 0.f; c[b * 5 + k] = 0.f; }
}

// ---------------------------------------------------------------------------
// One decoder LSTM step (input = concat(h_inp, a), hidden 5).
// ---------------------------------------------------------------------------
__global__ void __launch_bounds__(256)
k_dec_lstm(const float* __restrict__ h_inp, const float* __restrict__ a,
           const float* __restrict__ Wih, const float* __restrict__ Whh,
           const float* __restrict__ bih, const float* __restrict__ bhh,
           float* __restrict__ h, float* __restrict__ c)
{
    __shared__ float sWih[140], sWhh[100], sb[20];
    int tid = threadIdx.x;
    if (tid < 140) sWih[tid] = Wih[tid];
    if (tid < 100) sWhh[tid] = Whh[tid];
    if (tid < 20)  sb[tid] = bih[tid] + bhh[tid];
    __syncthreads();

    int b = blockIdx.x * 256 + tid;
    float inp[7];
#pragma unroll
    for (int k = 0; k < 5; ++k) inp[k] = h_inp[b * 5 + k];
    inp[5] = a[b * 2]; inp[6] = a[b * 2 + 1];
    float hv[5], cv[5];
#pragma unroll
    for (int k = 0; k < 5; ++k) { hv[k] = h[b * 5 + k]; cv[k] = c[b * 5 + k]; }

    float gates[20];
#pragma unroll
    for (int j = 0; j < 20; ++j) {
        float s = sb[j];
#pragma unroll
        for (int k = 0; k < 7; ++k) s = fmaf(inp[k], sWih[j * 7 + k], s);
#pragma unroll
        for (int k = 0; k < 5; ++k) s = fmaf(hv[k], sWhh[j * 5 + k], s);
        gates[j] = s;
    }
#pragma unroll
    for (int k = 0; k < 5; ++k) {
        float ig = sigm_(gates[k]);
        float fg = sigm_(gates[5 + k]);
        float gg = tanhf(gates[10 + k]);
        float og = sigm_(gates[15 + k]);
        cv[k] = fmaf(fg, cv[k], ig * gg);
        hv[k] = og * tanhf(cv[k]);
        h[b * 5 + k] = hv[k];
        c[b * 5 + k] = cv[k];
    }
}

// ---------------------------------------------------------------------------
// Sampling + output writes for one decoder step.
// ---------------------------------------------------------------------------
__global__ void __launch_bounds__(256)
k_sample(const float* __restrict__ mus, const float* __restrict__ lsig,
         const float* __restrict__ corr, const float* __restrict__ z,
         float* __restrict__ a, float* __restrict__ vpred,
         float* __restrict__ apred, int t)
{
    int b = blockIdx.x * 256 + threadIdx.x;
    float mu0 = mus[b * 2], mu1 = mus[b * 2 + 1];
    float l0 = lsig[b * 2], l1 = lsig[b * 2 + 1];
    float cr = corr[b];
    float rho = tanhf(cr);
    float s0 = __expf(l0), s1 = __expf(l1);
    float z0 = z[b * 2], z1 = z[b * 2 + 1];
    float x0 = fmaf(s0, z0, mu0);
    float x1 = fmaf(s1, fmaf(rho, z0, sqrtf(fmaxf(1.f - rho * rho, 0.f)) * z1), mu1);
    a[b * 2] = x0; a[b * 2 + 1] = x1;
    float* vp = vpred + b * (PREDL * 5) + t * 5;
    vp[0] = mu0; vp[1] = mu1; vp[2] = l0; vp[3] = l1; vp[4] = cr;
    apred[b * 2] = x0; apred[b * 2 + 1] = x1;
}

// ---------------------------------------------------------------------------
// Host orchestration
// ---------------------------------------------------------------------------
struct Lin { const float* W; const float* b; };
struct Mlp3 { Lin l[3]; };
struct LstmP { const float* Wih; const float* Whh; const float* bih; const float* bbh; };

extern "C" void kernel_launch(void* const* d_in, const int* in_sizes, int n_in,
                              void* d_out, int out_size, void* d_ws, size_t ws_size,
                              hipStream_t stream)
{
    (void)n_in; (void)out_size; (void)ws_size;
    auto F = [&](int i) { return (const float*)d_in[i]; };
    const float* v     = F(0);
    const float* maskp = F(1);
    const float* noise = F(2);

    Mlp3 traj, mus, sigm, corr, theta, phi, g;
    LstmP enc, dec;
    const float *p_mus, *p_sigma, *p_corr;
    auto M3 = [&](int i) { Mlp3 m = {{{F(i), F(i + 1)}, {F(i + 2), F(i + 3)}, {F(i + 4), F(i + 5)}}}; return m; };

    if (in_sizes[3] == 1024) {
        // dict-insertion-order flattening: traj_enc, enc_lstm, dec_lstm,
        // out_mus, out_sigma, out_corr, p_mus, p_sigma, p_corr, theta, phi, g
        traj = M3(3);
        enc  = {F(9), F(10), F(11), F(12)};
        dec  = {F(13), F(14), F(15), F(16)};
        mus  = M3(17); sigm = M3(23); corr = M3(29);
        p_mus = F(35); p_sigma = F(36); p_corr = F(37);
        theta = M3(38); phi = M3(44); g = M3(50);
    } else {
        // JAX sorted-key pytree order: dec_lstm{Whh,Wih,bhh,bih}, enc_lstm{...},
        // g, out_corr, out_mus, out_sigma, p_corr, p_mus, p_sigma, phi, theta, traj_enc
        dec  = {F(4), F(3), F(6), F(5)};
        enc  = {F(8), F(7), F(10), F(9)};
        g    = M3(11); corr = M3(17); mus = M3(23); sigm = M3(29);
        p_corr = F(35); p_mus = F(36); p_sigma = F(37);
        phi = M3(38); theta = M3(44); traj = M3(50);
    }

    // workspace layout (floats)
    float* W      = (float*)d_ws;
    float* H2     = W;                        // 32768*256
    float* feats  = H2 + 32768 * 256;         // 32768*5
    float* hinp   = feats + 32768 * 5;        // 4096*5
    float* tmp1   = hinp + NB * 5;            // 4096*128
    float* tmp2   = tmp1 + NB * 128;          // 4096*64
    float* thetaB = tmp2 + NB * 64;           // 4096*5
    float* phiB   = thetaB + NB * 5;
    float* gB     = phiB + NB * 5;
    float* dh     = gB + NB * 5;              // 4096*5
    float* dc     = dh + NB * 5;
    float* aB     = dc + NB * 5;              // 4096*2
    float* musB   = aB + NB * 2;              // 4096*2
    float* sigB   = musB + NB * 2;            // 4096*2
    float* corB   = sigB + NB * 2;            // 4096

    auto gemm = [&](const float* A, const float* Wt, const float* bias,
                    const float* slope, float* Cout, int M, int N, int K, int act) {
        int ntm = M >> 4, ntn = (N + 15) >> 4;
        int blocks = (ntm >> 3) * ntn;
        k_gemm<<<blocks, 256, 0, stream>>>(A, Wt, bias, slope, Cout, M, N, K, act);
    };

    // 1) traj_enc layers 1+2 fused (WMMA), then layer 3 (WMMA)
    k_enc12<<<4096, 256, 0, stream>>>(v, traj.l[0].W, traj.l[0].b,
                                      traj.l[1].W, traj.l[1].b, H2);
    gemm(H2, traj.l[2].W, traj.l[2].b, nullptr, feats, 32768, 5, 256, 0);

    // 2) encoder LSTM -> h_inp
    k_enc_lstm<<<NB / 256, 256, 0, stream>>>(feats, enc.Wih, enc.Whh, enc.bih, enc.bbh, hinp);

    // 3) non-local loop
    for (int it = 0; it < 3; ++it) {
        gemm(hinp, theta.l[0].W, theta.l[0].b, nullptr, tmp1, NB, 128, 5, 1);
        gemm(tmp1, theta.l[1].W, theta.l[1].b, nullptr, tmp2, NB, 64, 128, 1);
        gemm(tmp2, theta.l[2].W, theta.l[2].b, nullptr, thetaB, NB, 5, 64, 0);

        gemm(hinp, phi.l[0].W, phi.l[0].b, nullptr, tmp1, NB, 128, 5, 1);
        gemm(tmp1, phi.l[1].W, phi.l[1].b, nullptr, tmp2, NB, 64, 128, 1);
        gemm(tmp2, phi.l[2].W, phi.l[2].b, nullptr, phiB, NB, 5, 64, 0);

        gemm(hinp, g.l[0].W, g.l[0].b, nullptr, tmp1, NB, 128, 5, 1);
        gemm(tmp1, g.l[1].W, g.l[1].b, nullptr, tmp2, NB, 64, 128, 1);
        gemm(tmp2, g.l[2].W, g.l[2].b, nullptr, gB, NB, 5, 64, 0);

        k_attention<<<NB, 256, 0, stream>>>(thetaB, phiB, gB, maskp, hinp);
    }

    // 4) decoder
    float* out   = (float*)d_out;
    float* apred = out + NB * PREDL * 5;   // a_pred region, (13, B, 2)
    k_dec_init<<<NB / 256, 256, 0, stream>>>(v, aB, dh, dc, apred);

    for (int t = 0; t < PREDL; ++t) {
        k_dec_lstm<<<NB / 256, 256, 0, stream>>>(hinp, aB, dec.Wih, dec.Whh,
                                                 dec.bih, dec.bbh, dh, dc);
        gemm(dh,   mus.l[0].W,  mus.l[0].b,  p_mus,   tmp1, NB, 64, 5, 1);
        gemm(tmp1, mus.l[1].W,  mus.l[1].b,  p_mus,   tmp2, NB, 32, 64, 1);
        gemm(tmp2, mus.l[2].W,  mus.l[2].b,  nullptr, musB, NB, 2, 32, 0);

        gemm(dh,   sigm.l[0].W, sigm.l[0].b, p_sigma, tmp1, NB, 64, 5, 1);
        gemm(tmp1, sigm.l[1].W, sigm.l[1].b, p_sigma, tmp2, NB, 32, 64, 1);
        gemm(tmp2, sigm.l[2].W, sigm.l[2].b, nullptr, sigB, NB, 2, 32, 0);

        gemm(dh,   corr.l[0].W, corr.l[0].b, p_corr,  tmp1, NB, 64, 5, 1);
        gemm(tmp1, corr.l[1].W, corr.l[1].b, p_corr,  tmp2, NB, 32, 64, 1);
        gemm(tmp2, corr.l[2].W, corr.l[2].b, nullptr, corB, NB, 1, 32, 0);

        k_sample<<<NB / 256, 256, 0, stream>>>(musB, sigB, corB,
                                               noise + t * NB * 2, aB,
                                               out, apred + (t + 1) * NB * 2, t);
    }
}